// OccGridAccelBatchedDynamic_Base_21242908246600
// MI455X (gfx1250) — compile-verified
//
#include <hip/hip_runtime.h>
#include <hip/hip_bf16.h>
#include <stdint.h>

#define NFRAMES 16   // reference F
#define RES     96   // reference R

__global__ __launch_bounds__(256)
void occ_gather_kernel(const float* __restrict__ pts,
                       const float* __restrict__ ts,
                       const float* __restrict__ occ,
                       const float* __restrict__ kf,
                       const int*   __restrict__ bidx,
                       float* __restrict__ out,
                       int n)
{
    __shared__ __align__(16) float s_kf[NFRAMES];

    // ---- Stage the 16-entry keyframe table into LDS via the CDNA5 async
    // global->LDS engine (tracked by ASYNCcnt). Lanes 0..15 each move 4B.
    if (threadIdx.x < NFRAMES) {
        unsigned lds_byte =
            (unsigned)(unsigned long long)(uintptr_t)(&s_kf[threadIdx.x]);
        unsigned long long gaddr =
            (unsigned long long)(uintptr_t)(kf + threadIdx.x);
        asm volatile("global_load_async_to_lds_b32 %0, %1, off"
                     :: "v"(lds_byte), "v"(gaddr) : "memory");
    }
#if __has_builtin(__builtin_amdgcn_s_wait_asynccnt)
    __builtin_amdgcn_s_wait_asynccnt(0);
#else
    asm volatile("s_wait_asynccnt 0" ::: "memory");
#endif
    __syncthreads();

    const int i = blockIdx.x * blockDim.x + threadIdx.x;
    if (i >= n) return;

    // ---- Streamed per-point inputs: non-temporal so they don't evict the
    // 226MB occupancy grid from the 192MB L2.
    const float* p = pts + 3ll * i;
    const float px = __builtin_nontemporal_load(p + 0);
    const float py = __builtin_nontemporal_load(p + 1);
    const float pz = __builtin_nontemporal_load(p + 2);
    const float t  = __builtin_nontemporal_load(ts + i);
    const int   b  = __builtin_nontemporal_load(bidx + i);

    // ---- gidx = clamp(floor((p*0.5+0.5)*RES), 0, RES-1) ; (p*0.5+0.5)*96 = fma(p,48,48)
    const float hs = 0.5f * (float)RES;
    int gx = (int)fminf(fmaxf(floorf(fmaf(px, hs, hs)), 0.f), (float)(RES - 1));
    int gy = (int)fminf(fmaxf(floorf(fmaf(py, hs, hs)), 0.f), (float)(RES - 1));
    int gz = (int)fminf(fmaxf(floorf(fmaf(pz, hs, hs)), 0.f), (float)(RES - 1));

    // ---- searchsorted(left): ins = count(kf[k] < t). Branchless, 4x ds_load_b128.
    int ins = 0;
    const float4* kf4 = (const float4*)s_kf;
#pragma unroll
    for (int k = 0; k < NFRAMES / 4; ++k) {
        float4 v = kf4[k];
        ins += (v.x < t) + (v.y < t) + (v.z < t) + (v.w < t);
    }
    int lo = ins - 1;
    lo = lo < 0 ? 0 : lo;
    int hi = ins > (NFRAMES - 1) ? (NFRAMES - 1) : ins;
    const float klo = s_kf[lo];
    const float khi = s_kf[hi];
    const int fidx = (fabsf(khi - t) < fabsf(t - klo)) ? hi : lo;

    // ---- Flat gather: occ[(b*F+fidx), gx, gy, gz]. Default (RT) temporal
    // hint so the grid stays resident in L2 across the dispatch.
    long long cell = (long long)(b * NFRAMES + fidx);
    cell = (cell * RES + gx);
    cell = (cell * RES + gy);
    cell = (cell * RES + gz);
    const float v = occ[cell];

    __builtin_nontemporal_store(v, &out[i]);
}

extern "C" void kernel_launch(void* const* d_in, const int* in_sizes, int n_in,
                              void* d_out, int out_size, void* d_ws, size_t ws_size,
                              hipStream_t stream) {
    const float* pts  = (const float*)d_in[0];
    const float* ts   = (const float*)d_in[1];
    const float* occ  = (const float*)d_in[2];
    const float* kf   = (const float*)d_in[3];
    const int*   bidx = (const int*)d_in[4];
    float* out = (float*)d_out;

    const int n = in_sizes[1];           // number of query points (== len(ts))
    const int threads = 256;             // 8 wave32s per block
    const int blocks = (n + threads - 1) / threads;

    hipLaunchKernelGGL(occ_gather_kernel, dim3(blocks), dim3(threads), 0, stream,
                       pts, ts, occ, kf, bidx, out, n);
}